// TreeToTreeAttentionBatch_11373073399811
// MI455X (gfx1250) — compile-verified
//
#include <hip/hip_runtime.h>
#include <hip/hip_bf16.h>
#include <math.h>

// Problem constants (from reference): B=128, N=4096, H=256.
#define BB 128
#define NN 4096
#define HH 256
#define NCHUNK 8          // chunks of N per batch element
#define CHUNK (NN / NCHUNK) // 512 rows per chunk
#define WAVES 8           // waves per workgroup in hot kernel
#define ROWS_PER_WAVE (CHUNK / WAVES) // 64

typedef __attribute__((ext_vector_type(2))) float v2f;
typedef __attribute__((ext_vector_type(4))) float v4f;
typedef __attribute__((ext_vector_type(8))) float v8f;

// ---------------------------------------------------------------------------
// Kernel 1: u[b,h] = sum_k dh[b,k] * W_ah[k,h]   (fp32 WMMA 16x16x4)
// One wave per 16x16 output tile. grid = (B/16)*(H/16) = 8*16 = 128 blocks.
// ---------------------------------------------------------------------------
__global__ __launch_bounds__(32) void k_u_gemm(const float* __restrict__ dh,
                                               const float* __restrict__ W_ah,
                                               float* __restrict__ u) {
  const int tile = blockIdx.x;
  const int tm = (tile >> 4) << 4;   // b-tile origin (8 tiles of 16)
  const int tn = (tile & 15) << 4;   // h-tile origin (16 tiles of 16)
  const int lane = threadIdx.x & 31;
  const int half = lane >> 4;        // 0: lanes 0-15, 1: lanes 16-31
  const int l16  = lane & 15;

  v8f acc = {};
  for (int k = 0; k < HH; k += 4) {
    const int kk = k + 2 * half;     // this lane-half holds K = kk, kk+1
    v2f a, b;
    // A: 16x4 tile of dh; lane l16 = row M, VGPR j holds K = kk + j
    a.x = dh[(tm + l16) * HH + kk];
    a.y = dh[(tm + l16) * HH + kk + 1];
    // B: 4x16 tile of W_ah (K x N); lane l16 = col N
    b.x = W_ah[kk * HH + (tn + l16)];
    b.y = W_ah[(kk + 1) * HH + (tn + l16)];
    acc = __builtin_amdgcn_wmma_f32_16x16x4_f32(false, a, false, b,
                                                (short)0, acc, false, false);
  }
  // C/D layout: VGPR r -> M = r + 8*half, N = l16
#pragma unroll
  for (int r = 0; r < 8; ++r) {
    u[(tm + r + 8 * half) * HH + (tn + l16)] = acc[r];
  }
}

// ---------------------------------------------------------------------------
// Kernel 2 (hot): single streaming pass over annotations with online softmax.
// grid = (NCHUNK, B), block = 256 threads (8 wave32).
// Each wave owns 64 rows; per row each lane loads 2 float4s (coalesced b128),
// reduces the dot across the wave, then updates (m, d, c[]) online, reusing
// the already-loaded row values for the context accumulation.
// ---------------------------------------------------------------------------
__global__ __launch_bounds__(256) void k_attn(const float* __restrict__ ann,
                                              const float* __restrict__ u,
                                              float* __restrict__ part_m,
                                              float* __restrict__ part_d,
                                              float* __restrict__ part_c) {
  const int b     = blockIdx.y;
  const int chunk = blockIdx.x;
  const int wave  = threadIdx.x >> 5;
  const int lane  = threadIdx.x & 31;

  // Per-lane slice of u[b,:]: h in {4*lane..4*lane+3} and {128+4*lane..+3}
  const v4f* u4 = (const v4f*)(u + (size_t)b * HH);
  const v4f ua = u4[lane];
  const v4f ub = u4[lane + 32];

  float m = -INFINITY, d = 0.0f;
  v4f ca = {0.f, 0.f, 0.f, 0.f};
  v4f cb = {0.f, 0.f, 0.f, 0.f};

  const int n0 = chunk * CHUNK + wave * ROWS_PER_WAVE;
  const float* base = ann + ((size_t)b * NN + n0) * HH;

  for (int i = 0; i < ROWS_PER_WAVE; ++i) {
    const v4f* row4 = (const v4f*)(base + (size_t)i * HH);
    const v4f va = __builtin_nontemporal_load(row4 + lane);
    const v4f vb = __builtin_nontemporal_load(row4 + lane + 32);

    float dot = va.x * ua.x + va.y * ua.y + va.z * ua.z + va.w * ua.w
              + vb.x * ub.x + vb.y * ub.y + vb.z * ub.z + vb.w * ub.w;
#pragma unroll
    for (int off = 16; off; off >>= 1) dot += __shfl_xor(dot, off, 32);

    const float mn    = fmaxf(m, dot);
    const float alpha = __expf(m - mn);    // rescale old accumulators
    const float w     = __expf(dot - mn);  // weight of this row
    d = d * alpha + w;
    ca.x = ca.x * alpha + w * va.x;  ca.y = ca.y * alpha + w * va.y;
    ca.z = ca.z * alpha + w * va.z;  ca.w = ca.w * alpha + w * va.w;
    cb.x = cb.x * alpha + w * vb.x;  cb.y = cb.y * alpha + w * vb.y;
    cb.z = cb.z * alpha + w * vb.z;  cb.w = cb.w * alpha + w * vb.w;
    m = mn;
  }

  // Combine the 8 waves of this workgroup through LDS.
  __shared__ float s_m[WAVES];
  __shared__ float s_d[WAVES];
  __shared__ float s_c[WAVES][HH];

  s_c[wave][4 * lane + 0]       = ca.x;
  s_c[wave][4 * lane + 1]       = ca.y;
  s_c[wave][4 * lane + 2]       = ca.z;
  s_c[wave][4 * lane + 3]       = ca.w;
  s_c[wave][128 + 4 * lane + 0] = cb.x;
  s_c[wave][128 + 4 * lane + 1] = cb.y;
  s_c[wave][128 + 4 * lane + 2] = cb.z;
  s_c[wave][128 + 4 * lane + 3] = cb.w;
  if (lane == 0) { s_m[wave] = m; s_d[wave] = d; }
  __syncthreads();

  float M = s_m[0];
#pragma unroll
  for (int w2 = 1; w2 < WAVES; ++w2) M = fmaxf(M, s_m[w2]);

  const int pidx = b * NCHUNK + chunk;
  // Thread t owns output component h = t.
  float csum = 0.0f;
#pragma unroll
  for (int w2 = 0; w2 < WAVES; ++w2)
    csum += s_c[w2][threadIdx.x] * __expf(s_m[w2] - M);
  part_c[(size_t)pidx * HH + threadIdx.x] = csum;

  if (threadIdx.x == 0) {
    float D = 0.0f;
#pragma unroll
    for (int w2 = 0; w2 < WAVES; ++w2) D += s_d[w2] * __expf(s_m[w2] - M);
    part_m[pidx] = M;
    part_d[pidx] = D;
  }
}

// ---------------------------------------------------------------------------
// Kernel 3: combine NCHUNK partials per b -> context[b,h].
// grid = B, block = 256 (thread t owns h = t).
// ---------------------------------------------------------------------------
__global__ __launch_bounds__(256) void k_combine(const float* __restrict__ part_m,
                                                 const float* __restrict__ part_d,
                                                 const float* __restrict__ part_c,
                                                 float* __restrict__ context) {
  const int b = blockIdx.x;
  const int t = threadIdx.x;

  float M = -INFINITY;
#pragma unroll
  for (int w = 0; w < NCHUNK; ++w) M = fmaxf(M, part_m[b * NCHUNK + w]);

  float D = 0.0f, c = 0.0f;
#pragma unroll
  for (int w = 0; w < NCHUNK; ++w) {
    const float s = __expf(part_m[b * NCHUNK + w] - M);
    D += part_d[b * NCHUNK + w] * s;
    c += part_c[(size_t)(b * NCHUNK + w) * HH + t] * s;
  }
  context[b * HH + t] = c / D;
}

// ---------------------------------------------------------------------------
// Kernel 4: et[b,h] = tanh( sum_k cat[b,k] * W_ps[h,k] + b_ps[h] ), K = 2H.
// cat = [dh | context]. fp32 WMMA 16x16x4, one wave per 16x16 tile.
// ---------------------------------------------------------------------------
__global__ __launch_bounds__(32) void k_out_gemm(const float* __restrict__ dh,
                                                 const float* __restrict__ context,
                                                 const float* __restrict__ W_ps,
                                                 const float* __restrict__ b_ps,
                                                 float* __restrict__ out) {
  const int tile = blockIdx.x;
  const int tm = (tile >> 4) << 4;
  const int tn = (tile & 15) << 4;
  const int lane = threadIdx.x & 31;
  const int half = lane >> 4;
  const int l16  = lane & 15;

  v8f acc = {};
  for (int k = 0; k < 2 * HH; k += 4) {
    const int kk = k + 2 * half;
    const int bb = tm + l16;
    const int h  = tn + l16;
    v2f a, bm;
    a.x = (kk < HH) ? dh[bb * HH + kk] : context[bb * HH + kk - HH];
    a.y = (kk + 1 < HH) ? dh[bb * HH + kk + 1] : context[bb * HH + kk + 1 - HH];
    bm.x = W_ps[h * (2 * HH) + kk];
    bm.y = W_ps[h * (2 * HH) + kk + 1];
    acc = __builtin_amdgcn_wmma_f32_16x16x4_f32(false, a, false, bm,
                                                (short)0, acc, false, false);
  }
#pragma unroll
  for (int r = 0; r < 8; ++r) {
    const int bb = tm + r + 8 * half;
    const int h  = tn + l16;
    out[bb * HH + h] = tanhf(acc[r] + b_ps[h]);
  }
}

// ---------------------------------------------------------------------------
extern "C" void kernel_launch(void* const* d_in, const int* in_sizes, int n_in,
                              void* d_out, int out_size, void* d_ws, size_t ws_size,
                              hipStream_t stream) {
  const float* dh    = (const float*)d_in[0];   // [B,H]
  const float* ann   = (const float*)d_in[1];   // [B,N,H]
  const float* W_ah  = (const float*)d_in[2];   // [H,H]
  // d_in[3] = b_ah: constant shift over n, cancels in softmax -> unused.
  const float* W_ps  = (const float*)d_in[4];   // [H,2H]
  const float* b_ps  = (const float*)d_in[5];   // [H]
  float* out = (float*)d_out;                   // [B,H]

  // Workspace layout (floats):
  float* ws      = (float*)d_ws;
  float* u       = ws;                          // B*H          = 32768
  float* part_m  = u + BB * HH;                 // B*NCHUNK     = 1024
  float* part_d  = part_m + BB * NCHUNK;        // B*NCHUNK     = 1024
  float* part_c  = part_d + BB * NCHUNK;        // B*NCHUNK*H   = 262144
  float* context = part_c + (size_t)BB * NCHUNK * HH; // B*H    = 32768
  (void)in_sizes; (void)n_in; (void)out_size; (void)ws_size;

  // 1) u = dh @ W_ah  (WMMA fp32)
  k_u_gemm<<<dim3((BB / 16) * (HH / 16)), dim3(32), 0, stream>>>(dh, W_ah, u);

  // 2) fused logits + online softmax + context partials (single ann pass)
  k_attn<<<dim3(NCHUNK, BB), dim3(256), 0, stream>>>(ann, u, part_m, part_d, part_c);

  // 3) combine partials -> context
  k_combine<<<dim3(BB), dim3(256), 0, stream>>>(part_m, part_d, part_c, context);

  // 4) et = tanh([dh|context] @ W_ps^T + b_ps)  (WMMA fp32)
  k_out_gemm<<<dim3((BB / 16) * (HH / 16)), dim3(32), 0, stream>>>(dh, context, W_ps, b_ps, out);
}